// GraphModule_40922448396573
// MI455X (gfx1250) — compile-verified
//
#include <hip/hip_runtime.h>
#include <hip/hip_bf16.h>

// ---------------------------------------------------------------------------
// Problem constants (fixed by the reference)
// ---------------------------------------------------------------------------
#define GG 2048   // graphs
#define HH 256    // hidden features
#define DD 256    // branch output dim

typedef __attribute__((ext_vector_type(2))) float v2f;
typedef __attribute__((ext_vector_type(8))) float v8f;

// ---------------------------------------------------------------------------
// Kernel 1: exclusive prefix sum of num_nodes[2048] -> offsets[2048]
// Single block, 256 threads, 8 elements each.
// ---------------------------------------------------------------------------
__global__ void scan_offsets_kernel(const int* __restrict__ nn,
                                    int* __restrict__ offs) {
    __shared__ int chunk[256];
    const int t = threadIdx.x;
    int s = 0;
#pragma unroll
    for (int i = 0; i < 8; ++i) s += nn[t * 8 + i];
    chunk[t] = s;
    __syncthreads();
    if (t == 0) {
        int run = 0;
#pragma unroll 4
        for (int i = 0; i < 256; ++i) { int v = chunk[i]; chunk[i] = run; run += v; }
    }
    __syncthreads();
    int run = chunk[t];
#pragma unroll
    for (int i = 0; i < 8; ++i) {
        int v = nn[t * 8 + i];
        offs[t * 8 + i] = run;
        run += v;
    }
}

// ---------------------------------------------------------------------------
// Kernel 2: segment mean pooling.  One block per graph, 256 threads = one
// H-channel each; every loop iteration streams one fully-coalesced 1KB row.
// This pass reads ~150MB and is the HBM-bound part of the whole problem.
// ---------------------------------------------------------------------------
__global__ void pool_mean_kernel(const float* __restrict__ feat,
                                 const int* __restrict__ nn,
                                 const int* __restrict__ offs,
                                 float* __restrict__ M) {
    const int g = blockIdx.x;
    const int t = threadIdx.x;                 // channel 0..255
    const int off = offs[g];
    const int cnt = nn[g];                     // 16..128, never zero
    const float* p = feat + (size_t)off * HH + t;

    float a0 = 0.f, a1 = 0.f, a2 = 0.f, a3 = 0.f;
    int n = 0;
    for (; n + 4 <= cnt; n += 4) {
        a0 += p[(size_t)(n + 0) * HH];
        a1 += p[(size_t)(n + 1) * HH];
        a2 += p[(size_t)(n + 2) * HH];
        a3 += p[(size_t)(n + 3) * HH];
    }
    for (; n < cnt; ++n) a0 += p[(size_t)n * HH];

    const float inv = 1.0f / (float)cnt;
    M[(size_t)g * HH + t] = (a0 + a1 + a2 + a3) * inv;
}

// ---------------------------------------------------------------------------
// Kernel 3: three fused GEMMs  out = M[2048,256] @ W[256,256] + bias
// using native fp32 WMMA (v_wmma_f32_16x16x4_f32).  One wave per 16x16 output
// tile; the A operand (pooled means, L2-resident) is shared by all three
// weight matrices, so each K-step issues 3 WMMAs on one A load.
//
// ISA register layouts (CDNA5 ISA):
//   A 16x4 f32 : lane<16 -> K = k,k+1 ; lane>=16 -> K = k+2,k+3 (2 VGPRs)
//   B 4x16 f32 : VGPR0 row (k+2*hi), VGPR1 row (k+1+2*hi), lane&15 = column
//   C/D 16x16  : VGPR j -> row j + 8*hi, lane&15 = column
// Bias is pre-loaded into C (D = A*B + C).
// ---------------------------------------------------------------------------
__global__ void gemm3_wmma_kernel(const float* __restrict__ M,
                                  const float* __restrict__ Wp, const float* __restrict__ bp,
                                  const float* __restrict__ Wr, const float* __restrict__ br,
                                  const float* __restrict__ Wk, const float* __restrict__ bk,
                                  float* __restrict__ out) {
    const int wave = threadIdx.x >> 5;           // 4 waves per block
    const int lane = threadIdx.x & 31;
    const int tile = blockIdx.x * 4 + wave;      // 0..2047 (exact fit, no guard:
                                                 // EXEC stays all-ones for WMMA)
    const int row0 = (tile >> 4) * 16;           // 128 row tiles
    const int col0 = (tile & 15) * 16;           // 16 col tiles
    const int lm = lane & 15;
    const int hi = lane >> 4;

    // Seed accumulators with the (column-only) bias.
    const float bpv = bp[col0 + lm];
    const float brv = br[col0 + lm];
    const float bkv = bk[col0 + lm];
    v8f cp, cr, ck;
#pragma unroll
    for (int j = 0; j < 8; ++j) { cp[j] = bpv; cr[j] = brv; ck[j] = bkv; }

    const float* arow = M  + (size_t)(row0 + lm) * HH + 2 * hi;   // per-lane A row
    const float* wpc  = Wp + (size_t)(2 * hi) * DD + col0 + lm;   // per-lane B cols
    const float* wrc  = Wr + (size_t)(2 * hi) * DD + col0 + lm;
    const float* wkc  = Wk + (size_t)(2 * hi) * DD + col0 + lm;

#pragma unroll 4
    for (int k = 0; k < HH; k += 4) {
        // A: contiguous float2 at (row, k + 2*hi)
        v2f a = *(const v2f*)(arow + k);

        v2f bbp, bbr, bbk;
        const size_t ko = (size_t)k * DD;
        bbp.x = wpc[ko];  bbp.y = wpc[ko + DD];
        bbr.x = wrc[ko];  bbr.y = wrc[ko + DD];
        bbk.x = wkc[ko];  bbk.y = wkc[ko + DD];

        // 8-arg form: (neg_a, A, neg_b, B, c_mod, C, reuse_a, reuse_b)
        cp = __builtin_amdgcn_wmma_f32_16x16x4_f32(false, a, false, bbp, (short)0, cp, false, false);
        cr = __builtin_amdgcn_wmma_f32_16x16x4_f32(false, a, false, bbr, (short)0, cr, false, false);
        ck = __builtin_amdgcn_wmma_f32_16x16x4_f32(false, a, false, bbk, (short)0, ck, false, false);
    }

    // Tuple order: (keys, p_queries, r_queries)
    const size_t GD = (size_t)GG * DD;
    float* okeys = out;
    float* opq   = out + GD;
    float* orq   = out + 2 * GD;
#pragma unroll
    for (int j = 0; j < 8; ++j) {
        const int row = row0 + j + 8 * hi;
        const size_t idx = (size_t)row * DD + col0 + lm;
        okeys[idx] = ck[j];
        opq[idx]   = cp[j];
        orq[idx]   = cr[j];
    }
}

// ---------------------------------------------------------------------------
// Host-side launcher
// ---------------------------------------------------------------------------
extern "C" void kernel_launch(void* const* d_in, const int* in_sizes, int n_in,
                              void* d_out, int out_size, void* d_ws, size_t ws_size,
                              hipStream_t stream) {
    const float* features = (const float*)d_in[0];
    const int*   num_nodes = (const int*)d_in[1];
    const float* Wp = (const float*)d_in[2];
    const float* bp = (const float*)d_in[3];
    const float* Wr = (const float*)d_in[4];
    const float* br = (const float*)d_in[5];
    const float* Wk = (const float*)d_in[6];
    const float* bk = (const float*)d_in[7];
    float* out = (float*)d_out;

    // Workspace layout: [0, 8KB) offsets(int32[2048]); [8KB, 8KB+2MB) M[2048,256] f32
    int*   offs = (int*)d_ws;
    float* M    = (float*)((char*)d_ws + 8192);

    scan_offsets_kernel<<<1, 256, 0, stream>>>(num_nodes, offs);
    pool_mean_kernel<<<GG, 256, 0, stream>>>(features, num_nodes, offs, M);
    gemm3_wmma_kernel<<<GG / 4, 128, 0, stream>>>(M, Wp, bp, Wr, br, Wk, bk, out);
    (void)in_sizes; (void)n_in; (void)out_size; (void)ws_size;
}